// L2RegressionAttention_62560493633827
// MI455X (gfx1250) — compile-verified
//
#include <hip/hip_runtime.h>

// ---------------------------------------------------------------------------
// L2 regression attention (fast-weight memory scan) for MI455X / gfx1250.
//   B=4, S=4096, D=1024, H=16, hd=64, lr=0.1
// Matrix math on V_WMMA_F32_16X16X4_F32 (native f32 WMMA path); scan staging
// via GLOBAL_LOAD_ASYNC_TO_LDS_B128 (ASYNCcnt) in 8-step chunks.
// ---------------------------------------------------------------------------

typedef float v2f __attribute__((ext_vector_type(2)));
typedef float v8f __attribute__((ext_vector_type(8)));

#define WMMA_F32X4(a, b, c) \
    __builtin_amdgcn_wmma_f32_16x16x4_f32(false, (a), false, (b), (short)0, (c), false, false)

static constexpr int Bb = 4;
static constexpr int Ss = 4096;
static constexpr int Dd = 1024;
static constexpr int Hh = 16;
static constexpr int HD = 64;   // head dim
static constexpr int CH = 8;    // scan steps staged per async chunk

// --- CDNA5 async memory->LDS copy (ASYNCcnt path), inline asm ---------------
__device__ __forceinline__ void async_ld_lds_b128(float* lds, const float* gptr) {
    // Low 32 bits of the generic LDS pointer == LDS byte offset (aperture
    // layout); hardware adds the wave's LDS_BASE.
    const uint32_t off = (uint32_t)(uintptr_t)lds;
    asm volatile("global_load_async_to_lds_b128 %0, %1, off"
                 :: "v"(off), "v"(gptr)
                 : "memory");
}
__device__ __forceinline__ void wait_async0() {
    asm volatile("s_wait_asynccnt 0" ::: "memory");
}

// ---------------------------------------------------------------------------
// GEMM: Y = A @ W^T, A:[16384,1024] row-major, W:[1024,1024] row-major (torch
// Linear weight layout, contraction over W's second index).
// One wave -> 16x64 output strip (4 accumulator tiles, A-fragment reused 4x).
// qkv_mode=1: scatter result into scan layout [s][h][b][d].
// qkv_mode=0: plain row-major [row, col] store (final projection -> d_out).
// ---------------------------------------------------------------------------
__device__ __forceinline__ void store_tile(float* __restrict__ Y, int qkv_mode,
                                           int r0, int half8, int col, v8f acc) {
#pragma unroll
    for (int i = 0; i < 8; ++i) {
        const int row = r0 + i + half8;
        const float val = acc[i];
        if (qkv_mode) {
            const int bb = row >> 12;      // row = b*4096 + s
            const int ss = row & 4095;
            const int hh = col >> 6;       // col = h*64 + d
            const int dd = col & 63;
            Y[(((size_t)ss * Hh + hh) * Bb + bb) * HD + dd] = val;
        } else {
            Y[(size_t)row * Dd + col] = val;
        }
    }
}

__global__ __launch_bounds__(256) void gemm_xwt(const float* __restrict__ A,
                                                const float* __restrict__ W,
                                                float* __restrict__ Y,
                                                int qkv_mode) {
    const int wave  = threadIdx.x >> 5;
    const int lane  = threadIdx.x & 31;
    const int lrow  = lane & 15;
    const int khalf = (lane >> 4) << 1;   // 0 or 2 (A/B k split across lane halves)
    const int half8 = (lane >> 4) << 3;   // 0 or 8 (C/D row split)

    const int wtile = blockIdx.x * 8 + wave;  // 0..16383
    const int tn4   = wtile & 15;             // group of 4 N-tiles
    const int tm    = wtile >> 4;             // 0..1023 M-tile
    const int r0    = tm * 16;
    const int c0    = tn4 * 64;

    const float* Ap = A + (size_t)(r0 + lrow) * Dd + khalf;
    const float* Wp = W + (size_t)(c0 + lrow) * Dd + khalf;

    v8f acc0 = {}, acc1 = {}, acc2 = {}, acc3 = {};
#pragma unroll 4
    for (int kc = 0; kc < Dd; kc += 4) {
        const v2f a  = *(const v2f*)(Ap + kc);
        const v2f b0 = *(const v2f*)(Wp + kc);
        const v2f b1 = *(const v2f*)(Wp + 16 * Dd + kc);
        const v2f b2 = *(const v2f*)(Wp + 32 * Dd + kc);
        const v2f b3 = *(const v2f*)(Wp + 48 * Dd + kc);
        acc0 = WMMA_F32X4(a, b0, acc0);
        acc1 = WMMA_F32X4(a, b1, acc1);
        acc2 = WMMA_F32X4(a, b2, acc2);
        acc3 = WMMA_F32X4(a, b3, acc3);
    }

    store_tile(Y, qkv_mode, r0, half8, c0 + 0 * 16 + lrow, acc0);
    store_tile(Y, qkv_mode, r0, half8, c0 + 1 * 16 + lrow, acc1);
    store_tile(Y, qkv_mode, r0, half8, c0 + 2 * 16 + lrow, acc2);
    store_tile(Y, qkv_mode, r0, half8, c0 + 3 * 16 + lrow, acc3);
}

// ---------------------------------------------------------------------------
// Serial fast-weight scan. One block per head (16 blocks x 128 threads).
// Wave w owns output-dim rows [16w, 16w+16) of M (64x64, f32):
//   - M held in WMMA accumulator registers (4 x v8f, C/D layout),
//   - mirrored in LDS for A-operand gathers (in-wave DS ordering only).
// q/k/v staged CH steps at a time via async global->LDS DMA (ASYNCcnt);
// inside a chunk the steps run barrier-free (per-wave state only).
// Per step t:
//   v_pred = M_{t-1} K_t^T   (16 WMMA, K=4 chunks over hd)
//   err    = (V_t - v_pred) * lr/B           (VALU, transposed via LDS)
//   M_t    = WMMA(err^T, K_t, C=M_{t-1})     (4 WMMA, K=4 == batch -- exact fit)
//   out    = M_t Q_t^T       (16 WMMA)
// ---------------------------------------------------------------------------
__global__ __launch_bounds__(128) void l2reg_scan(const float* __restrict__ Qg,
                                                  const float* __restrict__ Kg,
                                                  const float* __restrict__ Vg,
                                                  float* __restrict__ Og) {
    __shared__ __align__(16) float Mlds[HD * HD];        // 16 KB, per-wave rows
    __shared__ __align__(16) float kst[CH][Bb * HD];     // 8 KB  K chunk
    __shared__ __align__(16) float qst[CH][Bb * HD];     // 8 KB  Q chunk
    __shared__ __align__(16) float vst[CH][Bb * HD];     // 8 KB  V chunk
    __shared__ __align__(16) float errbuf[4][16 * 4];    // per-wave err^T [d][b]

    const int h     = blockIdx.x;
    const int tid   = threadIdx.x;
    const int wave  = tid >> 5;
    const int lane  = tid & 31;
    const int dw    = wave << 4;        // this wave's d-row block
    const int lrow  = lane & 15;
    const int khalf = (lane >> 4) << 1; // 0/2
    const int half8 = (lane >> 4) << 3; // 0/8
    const int bcol  = lrow & 3;         // clamp batch column (cols 4..15 unused)

    for (int i = tid; i < HD * HD; i += 128) Mlds[i] = 0.0f;

    v8f M0 = {}, M1 = {}, M2 = {}, M3 = {};
    const float lr_over_b = 0.1f / 4.0f;

    for (int c = 0; c < Ss / CH; ++c) {
        const int t0 = c * CH;

        // ---- async-stage CH steps of q/k/v: 3 x 2048 floats, b128 DMA ----
#pragma unroll
        for (int u = 0; u < 4; ++u) {
            const int idx = u * 128 + tid;     // 0..511 b128 transfers per array
            const int f0  = idx << 2;          // float offset 0..2044
            const int j   = f0 >> 8;           // step within chunk
            const int w   = f0 & 255;          // offset within [b][d] block
            const size_t g = (((size_t)(t0 + j) * Hh + h) * (Bb * HD)) + w;
            async_ld_lds_b128(&kst[0][0] + f0, Kg + g);
            async_ld_lds_b128(&qst[0][0] + f0, Qg + g);
            async_ld_lds_b128(&vst[0][0] + f0, Vg + g);
        }
        wait_async0();
        __syncthreads();

        for (int j = 0; j < CH; ++j) {
            const int t = t0 + j;
            const float* kb = kst[j];
            const float* qb = qst[j];
            const float* vb = vst[j];

            // ---- v_pred tile: rows = d (16), cols = batch (4 valid of 16) ----
            v8f vp = {};
#pragma unroll
            for (int kc = 0; kc < HD; kc += 4) {
                const v2f a = *(const v2f*)&Mlds[(dw + lrow) * HD + kc + khalf];
                const v2f b = *(const v2f*)&kb[bcol * HD + kc + khalf];
                vp = WMMA_F32X4(a, b, vp);
            }

            // ---- scaled error, transposed into A-layout scratch ----
            if (lrow < 4) {
#pragma unroll
                for (int i = 0; i < 8; ++i) {
                    const float v = vb[lrow * HD + dw + half8 + i];
                    errbuf[wave][(i + half8) * 4 + lrow] = (v - vp[i]) * lr_over_b;
                }
            }

            // ---- M update: M_j += err^T(16x4) @ K_t(4x16), contraction = batch ----
            const v2f ef = *(const v2f*)&errbuf[wave][lrow * 4 + khalf];
            {
                v2f b;
                b.x = kb[(khalf + 0) * HD + 0 * 16 + lrow];
                b.y = kb[(khalf + 1) * HD + 0 * 16 + lrow];
                M0 = WMMA_F32X4(ef, b, M0);
                b.x = kb[(khalf + 0) * HD + 1 * 16 + lrow];
                b.y = kb[(khalf + 1) * HD + 1 * 16 + lrow];
                M1 = WMMA_F32X4(ef, b, M1);
                b.x = kb[(khalf + 0) * HD + 2 * 16 + lrow];
                b.y = kb[(khalf + 1) * HD + 2 * 16 + lrow];
                M2 = WMMA_F32X4(ef, b, M2);
                b.x = kb[(khalf + 0) * HD + 3 * 16 + lrow];
                b.y = kb[(khalf + 1) * HD + 3 * 16 + lrow];
                M3 = WMMA_F32X4(ef, b, M3);
            }

            // ---- write updated M back to LDS mirror (own rows only) ----
#pragma unroll
            for (int i = 0; i < 8; ++i) {
                const int rbase = (dw + half8 + i) * HD + lrow;
                Mlds[rbase + 0 * 16] = M0[i];
                Mlds[rbase + 1 * 16] = M1[i];
                Mlds[rbase + 2 * 16] = M2[i];
                Mlds[rbase + 3 * 16] = M3[i];
            }

            // ---- out tile: M_t @ Q_t^T ----
            v8f ot = {};
#pragma unroll
            for (int kc = 0; kc < HD; kc += 4) {
                const v2f a = *(const v2f*)&Mlds[(dw + lrow) * HD + kc + khalf];
                const v2f b = *(const v2f*)&qb[bcol * HD + kc + khalf];
                ot = WMMA_F32X4(a, b, ot);
            }

            // ---- store out (layout [b*S+s, D] for the output projection) ----
            if (lrow < 4) {
                const size_t ob = (((size_t)lrow * Ss + t) * Dd) + h * HD + dw + half8;
#pragma unroll
                for (int i = 0; i < 8; ++i) Og[ob + i] = ot[i];
            }
        }
        __syncthreads();
    }
}

// ---------------------------------------------------------------------------
extern "C" void kernel_launch(void* const* d_in, const int* in_sizes, int n_in,
                              void* d_out, int out_size, void* d_ws, size_t ws_size,
                              hipStream_t stream) {
    (void)in_sizes; (void)n_in; (void)out_size; (void)ws_size;

    const float* x  = (const float*)d_in[0];
    const float* Wq = (const float*)d_in[1];
    const float* Wk = (const float*)d_in[2];
    const float* Wv = (const float*)d_in[3];
    const float* Wo = (const float*)d_in[4];
    float* out = (float*)d_out;

    const size_t NE = (size_t)Bb * Ss * Dd;  // 16,777,216 floats = 64 MB
    float* Qw = (float*)d_ws;                // scan layout [s][h][b][d]
    float* Kw = Qw + NE;
    float* Vw = Kw + NE;
    float* Aw = Vw + NE;                     // attention out [b*S+s][D]

    const dim3 gblk(256);
    const dim3 ggrid(2048);  // 2048 blocks * 8 waves = 1024 M-tiles * 16 N-groups

    gemm_xwt<<<ggrid, gblk, 0, stream>>>(x, Wq, Qw, 1);
    gemm_xwt<<<ggrid, gblk, 0, stream>>>(x, Wk, Kw, 1);
    gemm_xwt<<<ggrid, gblk, 0, stream>>>(x, Wv, Vw, 1);

    l2reg_scan<<<dim3(16), dim3(128), 0, stream>>>(Qw, Kw, Vw, Aw);

    gemm_xwt<<<ggrid, gblk, 0, stream>>>(Aw, Wo, out, 0);
}